// GCNEncoder_56616258895907
// MI455X (gfx1250) — compile-verified
//
#include <hip/hip_runtime.h>
#include <hip/hip_bf16.h>

// ---------------------------------------------------------------------------
// GCN (2-layer, PyG GCNConv semantics) for MI455X / gfx1250.
//   N=50000 nodes, E=800000 edges, C_in=256, C_hid=256, C_out=128.
//   Dense GEMMs via V_WMMA_F32_16X16X4_F32 (exact fp32 tensor path).
//   Aggregation via float4 gather + global_atomic_add_f32 scatter (L2-bound).
// ---------------------------------------------------------------------------

typedef __attribute__((ext_vector_type(2))) float f32x2;
typedef __attribute__((ext_vector_type(8))) float f32x8;

#define N_NODES 50000
#define N_EDGES 800000

// ---------------------------------------------------------------------------
// Degree: deg[dst] += 1 per edge (self loops added later as +1 in rsqrt).
// ---------------------------------------------------------------------------
__global__ __launch_bounds__(256) void degree_kernel(
    const long long* __restrict__ dst, float* __restrict__ deg, int E) {
  int i = blockIdx.x * 256 + threadIdx.x;
  if (i < E) {
    atomicAdd(&deg[(int)dst[i]], 1.0f);
  }
}

// dinv[i] = rsqrt(deg[i] + 1)   (+1 = self loop; guarantees deg > 0)
__global__ __launch_bounds__(256) void dinv_kernel(float* __restrict__ deg, int N) {
  int i = blockIdx.x * 256 + threadIdx.x;
  if (i < N) {
    deg[i] = rsqrtf(deg[i] + 1.0f);
  }
}

// ---------------------------------------------------------------------------
// WMMA fp32 GEMM:  C[M x NC] = A[M x 256] * B[256 x NC]
//   grid.x = M/16 tiles; block = 32 * (NC/16) threads (one wave per 16-col
//   stripe). A tile staged in LDS (padded to kill bank conflicts); B read
//   straight from L2 (W1/W2 are 256/128 KB, fully L2-resident).
// ---------------------------------------------------------------------------
template <int NC>
__global__ __launch_bounds__(32 * (NC / 16)) void gemm_wmma_f32(
    const float* __restrict__ A, const float* __restrict__ B,
    float* __restrict__ C, int M) {
  constexpr int K = 256;
  constexpr int LDSS = K + 4;  // pad 4 floats: row stride 260 -> 4-bank skew
  constexpr int NTHREADS = 32 * (NC / 16);
  __shared__ float As[16 * LDSS];

  const int tid = threadIdx.x;
  const int m0 = blockIdx.x * 16;

  // Cooperative load of the 16x256 A tile (float4 chunks).
  for (int i = tid; i < 16 * (K / 4); i += NTHREADS) {
    const int r = i / (K / 4);
    const int c4 = i % (K / 4);
    const float4 v =
        reinterpret_cast<const float4*>(A + (size_t)(m0 + r) * K)[c4];
    float* p = &As[r * LDSS + c4 * 4];
    p[0] = v.x; p[1] = v.y; p[2] = v.z; p[3] = v.w;
  }
  __syncthreads();

  const int lane = tid & 31;
  const int wave = tid >> 5;
  const int n0 = wave * 16;
  const int mf = lane & 15;          // A-frag row / B-frag col / C col
  const int khalf = (lane >> 4) * 2; // 0 for lanes 0-15, 2 for lanes 16-31

  f32x8 acc = {};
#pragma unroll
  for (int k0 = 0; k0 < K; k0 += 4) {
    const int kb = k0 + khalf;
    // A fragment: lane l holds A[mf][kb], A[mf][kb+1]  (8B-aligned in LDS)
    const f32x2 a = *reinterpret_cast<const f32x2*>(&As[mf * LDSS + kb]);
    // B fragment: lane l holds B[kb][n0+mf], B[kb+1][n0+mf]
    f32x2 b;
    b.x = B[(size_t)kb * NC + n0 + mf];
    b.y = B[(size_t)(kb + 1) * NC + n0 + mf];
    acc = __builtin_amdgcn_wmma_f32_16x16x4_f32(
        /*neg_a=*/false, a, /*neg_b=*/false, b,
        /*c_mod=*/(short)0, acc, /*reuse_a=*/false, /*reuse_b=*/false);
  }

  // Store D: VGPR r -> row (m0 + r + 8*(lane>>4)), col (n0 + (lane&15)).
  const int col = n0 + mf;
  const int mbase = m0 + ((lane >> 4) * 8);
#pragma unroll
  for (int r = 0; r < 8; ++r) {
    C[(size_t)(mbase + r) * NC + col] = acc[r];
  }
}

// ---------------------------------------------------------------------------
// Edge scatter:  AGG[dst] += H[src] * dinv[src]*dinv[dst]   (float4 + atomics)
//   block = 256 threads -> (256 / (C/4)) edges per block.
// ---------------------------------------------------------------------------
template <int C>
__global__ __launch_bounds__(256) void scatter_kernel(
    const long long* __restrict__ src, const long long* __restrict__ dst,
    const float* __restrict__ dinv, const float* __restrict__ H,
    float* __restrict__ AGG, int E) {
  constexpr int C4 = C / 4;
  constexpr int EPB = 256 / C4;
  const int t = threadIdx.x;
  const int e = blockIdx.x * EPB + t / C4;
  if (e >= E) return;
  const int c = (t % C4) * 4;
  const int s = (int)src[e];
  const int d = (int)dst[e];
  const float nrm = dinv[s] * dinv[d];
  const float4 v =
      *reinterpret_cast<const float4*>(H + (size_t)s * C + c);
  float* o = AGG + (size_t)d * C + c;
  atomicAdd(o + 0, v.x * nrm);
  atomicAdd(o + 1, v.y * nrm);
  atomicAdd(o + 2, v.z * nrm);
  atomicAdd(o + 3, v.w * nrm);
}

// ---------------------------------------------------------------------------
// Fused epilogue (in place on AGG):
//   AGG[i] = (AGG[i] + H[i]*dinv[i]^2 + bias)   [ReLU optional]
//   (H[i]*dinv[i]^2 is the self-loop message; atomic-free elementwise pass.)
// ---------------------------------------------------------------------------
template <int C, bool RELU>
__global__ __launch_bounds__(256) void finalize_kernel(
    float* __restrict__ AGG, const float* __restrict__ H,
    const float* __restrict__ bias, const float* __restrict__ dinv, int N) {
  constexpr int C4 = C / 4;
  constexpr int NPB = 256 / C4;
  const int t = threadIdx.x;
  const int n = blockIdx.x * NPB + t / C4;
  if (n >= N) return;
  const int c = (t % C4) * 4;
  const float d2 = dinv[n] * dinv[n];
  const float4 a = *reinterpret_cast<const float4*>(AGG + (size_t)n * C + c);
  const float4 h = *reinterpret_cast<const float4*>(H + (size_t)n * C + c);
  const float4 b = *reinterpret_cast<const float4*>(bias + c);
  float4 r;
  r.x = a.x + h.x * d2 + b.x;
  r.y = a.y + h.y * d2 + b.y;
  r.z = a.z + h.z * d2 + b.z;
  r.w = a.w + h.w * d2 + b.w;
  if (RELU) {
    r.x = fmaxf(r.x, 0.0f);
    r.y = fmaxf(r.y, 0.0f);
    r.z = fmaxf(r.z, 0.0f);
    r.w = fmaxf(r.w, 0.0f);
  }
  *reinterpret_cast<float4*>(AGG + (size_t)n * C + c) = r;
}

// ---------------------------------------------------------------------------
static inline size_t align256(size_t x) { return (x + 255) & ~(size_t)255; }

extern "C" void kernel_launch(void* const* d_in, const int* in_sizes, int n_in,
                              void* d_out, int out_size, void* d_ws,
                              size_t ws_size, hipStream_t stream) {
  const float* x = (const float*)d_in[0];
  const long long* ei = (const long long*)d_in[1];  // int64 [2, E]
  const float* W1 = (const float*)d_in[2];
  const float* b1 = (const float*)d_in[3];
  const float* W2 = (const float*)d_in[4];
  const float* b2 = (const float*)d_in[5];
  float* out = (float*)d_out;

  const int N = N_NODES;
  const int E = N_EDGES;
  constexpr int INC = 256, HID = 256, OUTC = 128;

  const long long* src = ei;      // edge_index[0]
  const long long* dst = ei + E;  // edge_index[1]

  // Workspace carve-up (deg/dinv reuse one buffer; h reused as h2).
  char* ws = (char*)d_ws;
  size_t o = 0;
  float* dinv = (float*)(ws + o); o += align256((size_t)N * sizeof(float));
  float* h    = (float*)(ws + o); o += align256((size_t)N * HID * sizeof(float));
  float* agg  = (float*)(ws + o); o += align256((size_t)N * HID * sizeof(float));
  (void)ws_size;

  // Zero accumulators (graph-capture-safe async memsets).
  hipMemsetAsync(dinv, 0, (size_t)N * sizeof(float), stream);
  hipMemsetAsync(agg, 0, (size_t)N * HID * sizeof(float), stream);
  hipMemsetAsync(out, 0, (size_t)N * OUTC * sizeof(float), stream);

  // Degree + symmetric normalization.
  degree_kernel<<<(E + 255) / 256, 256, 0, stream>>>(dst, dinv, E);
  dinv_kernel<<<(N + 255) / 256, 256, 0, stream>>>(dinv, N);

  // ---- Layer 1: h = x @ W1 ; agg = scatter(norm * h[src]) ; relu(+b1+self)
  gemm_wmma_f32<HID><<<N / 16, 32 * (HID / 16), 0, stream>>>(x, W1, h, N);
  scatter_kernel<HID><<<(E * (HID / 4) + 255) / 256, 256, 0, stream>>>(
      src, dst, dinv, h, agg, E);
  finalize_kernel<HID, true><<<(N * (HID / 4) + 255) / 256, 256, 0, stream>>>(
      agg, h, b1, dinv, N);  // agg now holds h1 = relu(...)

  // ---- Layer 2: h2 = h1 @ W2 ; out = scatter(norm * h2[src]) + self + b2
  gemm_wmma_f32<OUTC><<<N / 16, 32 * (OUTC / 16), 0, stream>>>(agg, W2, h, N);
  scatter_kernel<OUTC><<<(E * (OUTC / 4) + 255) / 256, 256, 0, stream>>>(
      src, dst, dinv, h, out, E);
  finalize_kernel<OUTC, false><<<(N * (OUTC / 4) + 255) / 256, 256, 0, stream>>>(
      out, h, b2, dinv, N);
}